// path_13855564497394
// MI455X (gfx1250) — compile-verified
//
#include <hip/hip_runtime.h>
#include <cmath>

// Rigid-body scan, fully parallelized via associative decomposition:
//   wz       : prefix sum of 0.5*tau_z
//   (wx,wy)  : 2x2 time-varying affine scan (given wz)
//   R        : prefix product of 3x3 Rodrigues matrices (chunked scan; the
//              "apply chunk base" phase uses V_WMMA_F32_16X16X4_F32, 4
//              independent 3x3 products per WMMA via 4-row/4-col banding)
//   v, pos   : single / double prefix sums of dv
// 160 chunks x 125 steps = 20000.

#define STEPS  20000
#define CHUNKS 160
#define CS     125
#define DTF    0.01f
#define GACC   9.81f

// workspace layout (float offsets)
#define OFF_SZZ    0        // 160   chunk sums of 0.5*tau_z
#define OFF_BWZ    160      // 160   wz at chunk start
#define OFF_AFF    320      // 160*6 chunk affine (A00,A01,A10,A11,d0,d1)
#define OFF_SXY    1280     // 160*2 (wx,wy) at chunk start
#define OFF_RBASE  1600     // 160*9 R state at chunk start
#define OFF_SUM1   3040     // 160*3 chunk sum of dv
#define OFF_SUM2   3520     // 160*3 chunk sum of local prefix sums
#define OFF_B1     4000     // 160*3
#define OFF_B2     4480     // 160*3
#define OFF_L      4960     // 20000*9 local inclusive E-products
// total 184960 floats = ~740 KB

typedef float v2f __attribute__((ext_vector_type(2)));
typedef float v8f __attribute__((ext_vector_type(8)));

__device__ __forceinline__ void mm3(float* __restrict__ d,
                                    const float* __restrict__ a,
                                    const float* __restrict__ b) {
#pragma unroll
  for (int r = 0; r < 3; ++r)
#pragma unroll
    for (int c = 0; c < 3; ++c)
      d[r*3+c] = a[r*3+0]*b[0*3+c] + a[r*3+1]*b[1*3+c] + a[r*3+2]*b[2*3+c];
}

// ---- pass-through of actions to output tail (vectorized b128 copies) ----
__global__ void k_copy_actions(const float4* __restrict__ act,
                               float4* __restrict__ dst, int n) {
  int i = blockIdx.x * blockDim.x + threadIdx.x;
  if (i < n) dst[i] = act[i];
}

// ---- wz: chunk partial sums ----
__global__ void k_chunk_wz(const float* __restrict__ act, float* __restrict__ ws) {
  int c = blockIdx.x * blockDim.x + threadIdx.x;
  if (c >= CHUNKS) return;
  float s = 0.f;
  for (int i = 0; i < CS; ++i) s += 0.5f * act[(c*CS + i)*4 + 3];
  ws[OFF_SZZ + c] = s;
}

__global__ void k_scan_wz(const float* __restrict__ x, float* __restrict__ ws) {
  float wz = x[17];
  for (int c = 0; c < CHUNKS; ++c) { ws[OFF_BWZ + c] = wz; wz += ws[OFF_SZZ + c]; }
}

// ---- (wx,wy): per-chunk affine composition ----
__global__ void k_chunk_aff(const float* __restrict__ act, float* __restrict__ ws) {
  int c = blockIdx.x * blockDim.x + threadIdx.x;
  if (c >= CHUNKS) return;
  float wz = ws[OFF_BWZ + c];
  float A00 = 1.f, A01 = 0.f, A10 = 0.f, A11 = 1.f, d0 = 0.f, d1 = 0.f;
  for (int i = 0; i < CS; ++i) {
    int t = c*CS + i;
    float tx = act[t*4+1], ty = act[t*4+2], tz = act[t*4+3];
    float k = 0.01f * wz;                       // DT*INV_I*(Izz-Ixx) folded
    float nA00 = A00 - k*A10, nA01 = A01 - k*A11;
    float nA10 = k*A00 + A10, nA11 = k*A01 + A11;
    float nd0  = d0 - k*d1 + tx;                // DT*100*tau = tau
    float nd1  = k*d0 + d1 + ty;
    A00 = nA00; A01 = nA01; A10 = nA10; A11 = nA11; d0 = nd0; d1 = nd1;
    wz += 0.5f * tz;
  }
  float* p = ws + OFF_AFF + c*6;
  p[0]=A00; p[1]=A01; p[2]=A10; p[3]=A11; p[4]=d0; p[5]=d1;
}

__global__ void k_scan_aff(const float* __restrict__ x, float* __restrict__ ws) {
  float wx = x[15], wy = x[16];
  for (int c = 0; c < CHUNKS; ++c) {
    ws[OFF_SXY + 2*c] = wx; ws[OFF_SXY + 2*c + 1] = wy;
    const float* p = ws + OFF_AFF + c*6;
    float nwx = p[0]*wx + p[1]*wy + p[4];
    float nwy = p[2]*wx + p[3]*wy + p[5];
    wx = nwx; wy = nwy;
  }
}

// ---- full omega sequence + Rodrigues E_t + local R prefixes ----
__global__ void k_omega_E(const float* __restrict__ act, float* __restrict__ ws,
                          float* __restrict__ out) {
  int c = blockIdx.x * blockDim.x + threadIdx.x;
  if (c >= CHUNKS) return;
  float wx = ws[OFF_SXY + 2*c], wy = ws[OFF_SXY + 2*c + 1], wz = ws[OFF_BWZ + c];
  float L[9], E[9], tmp[9];
  for (int i = 0; i < CS; ++i) {
    int t = c*CS + i;
    // E_t = exp(skew(omega_t * DT))
    float ax = wx*DTF, ay = wy*DTF, az = wz*DTF;
    float th = sqrtf(ax*ax + ay*ay + az*az);
    if (th > 0.f) {
      float inv = 1.f/th, nx = ax*inv, ny = ay*inv, nz = az*inv;
      float s = sinf(th), cmc = 1.f - cosf(th);
      E[0]=1.f-cmc*(ny*ny+nz*nz); E[1]=-s*nz+cmc*nx*ny;        E[2]= s*ny+cmc*nx*nz;
      E[3]= s*nz+cmc*nx*ny;       E[4]=1.f-cmc*(nx*nx+nz*nz);  E[5]=-s*nx+cmc*ny*nz;
      E[6]=-s*ny+cmc*nx*nz;       E[7]= s*nx+cmc*ny*nz;        E[8]=1.f-cmc*(nx*nx+ny*ny);
    } else {
      E[0]=1;E[1]=0;E[2]=0;E[3]=0;E[4]=1;E[5]=0;E[6]=0;E[7]=0;E[8]=1;
    }
    if (i == 0) { for (int k=0;k<9;++k) L[k]=E[k]; }
    else        { mm3(tmp, L, E); for (int k=0;k<9;++k) L[k]=tmp[k]; }
    float* Lp = ws + OFF_L + (size_t)t*9;
    for (int k=0;k<9;++k) Lp[k] = L[k];
    // omega_{t+1}
    float tx = act[t*4+1], ty = act[t*4+2], tz = act[t*4+3];
    float nwx = wx + tx - 0.01f*wy*wz;
    float nwy = wy + ty + 0.01f*wx*wz;
    float nwz = wz + 0.5f*tz;
    out[t*18+15] = nwx; out[t*18+16] = nwy; out[t*18+17] = nwz;
    wx = nwx; wy = nwy; wz = nwz;
  }
}

// ---- top-level scan of chunk R products (R0 folded in) ----
__global__ void k_scan_R(const float* __restrict__ x, float* __restrict__ ws) {
  float Rb[9], tmp[9];
  for (int k=0;k<9;++k) Rb[k] = x[6+k];
  for (int c = 0; c < CHUNKS; ++c) {
    float* p = ws + OFF_RBASE + c*9;
    for (int k=0;k<9;++k) p[k] = Rb[k];
    const float* T = ws + OFF_L + ((size_t)(c*CS + CS - 1))*9;  // chunk total
    mm3(tmp, Rb, T);
    for (int k=0;k<9;++k) Rb[k] = tmp[k];
  }
}

// ---- apply: states[g].R = Rbase[g/CS] @ L[g], 4 products per WMMA ----
// A(16x4): 4-row bands hold Rbase_j rows (padded); B(4x16): 4-col bands hold
// L_j cols (row K=3 zeroed). Diagonal 4x4 blocks of D are the 3x3 products.
__global__ void k_wmma_apply(const float* __restrict__ ws, float* __restrict__ out) {
  __shared__ float tile[256];
  int l = threadIdx.x;
  int gbase = blockIdx.x * 4;
  int half = l >> 4, lm = l & 15;
  int j = lm >> 2, rc = lm & 3;               // band index, row/col within band
  // A operand: lane half selects K={0,1} vs {2,3}; VGPR index is K within half
  const float* Rb = ws + OFF_RBASE + ((gbase + j) / CS) * 9;
  int k0 = 2*half, k1 = 2*half + 1;
  v2f a;
  a.x = (rc < 3 && k0 < 3) ? Rb[rc*3 + k0] : 0.f;
  a.y = (rc < 3 && k1 < 3) ? Rb[rc*3 + k1] : 0.f;
  // B operand (4x16): assume row K = 2*half + vgpr (mirrors A striping)
  const float* Lp = ws + OFF_L + (size_t)(gbase + j) * 9;
  v2f b;
  b.x = (k0 < 3 && rc < 3) ? Lp[k0*3 + rc] : 0.f;
  b.y = (k1 < 3 && rc < 3) ? Lp[k1*3 + rc] : 0.f;
  v8f acc = {};
  acc = __builtin_amdgcn_wmma_f32_16x16x4_f32(false, a, false, b,
                                              (short)0, acc, false, false);
#pragma unroll
  for (int v = 0; v < 8; ++v) {               // D: lane half -> M = v + 8*half
    int m = v + 8*half;
    tile[m*16 + lm] = acc[v];
  }
  __syncthreads();
  for (int tsk = l; tsk < 36; tsk += 32) {    // 4 products x 9 elements
    int pj = tsk / 9, e = tsk % 9, r = e / 3, q = e % 3;
    out[(gbase + pj)*18 + 6 + e] = tile[(4*pj + r)*16 + (4*pj + q)];
  }
}

// ---- dv chunk aggregates (reads R column 2 from already-written output) ----
__device__ __forceinline__ void get_col2(const float* __restrict__ x,
                                         const float* __restrict__ out, int t,
                                         float& c0, float& c1, float& c2) {
  if (t == 0) { c0 = x[8]; c1 = x[11]; c2 = x[14]; }
  else {
    const float* Rp = out + (t-1)*18 + 6;
    c0 = Rp[2]; c1 = Rp[5]; c2 = Rp[8];
  }
}

__global__ void k_dv(const float* __restrict__ x, const float* __restrict__ act,
                     const float* __restrict__ out, float* __restrict__ ws) {
  int c = blockIdx.x * blockDim.x + threadIdx.x;
  if (c >= CHUNKS) return;
  float s1x=0,s1y=0,s1z=0, s2x=0,s2y=0,s2z=0;
  for (int i = 0; i < CS; ++i) {
    int t = c*CS + i;
    float c0,c1,c2; get_col2(x, out, t, c0, c1, c2);
    float fz = act[t*4];
    float dx = fz*c0, dy = fz*c1, dz = fz*c2 - GACC;
    s1x += dx; s1y += dy; s1z += dz;
    s2x += s1x; s2y += s1y; s2z += s1z;
  }
  float* p1 = ws + OFF_SUM1 + c*3; p1[0]=s1x; p1[1]=s1y; p1[2]=s1z;
  float* p2 = ws + OFF_SUM2 + c*3; p2[0]=s2x; p2[1]=s2y; p2[2]=s2z;
}

__global__ void k_scan_vp(float* __restrict__ ws) {
  float B1[3] = {0,0,0}, B2[3] = {0,0,0};
  for (int c = 0; c < CHUNKS; ++c) {
    for (int k=0;k<3;++k) { ws[OFF_B1+c*3+k]=B1[k]; ws[OFF_B2+c*3+k]=B2[k]; }
    const float* s1 = ws + OFF_SUM1 + c*3;
    const float* s2 = ws + OFF_SUM2 + c*3;
    for (int k=0;k<3;++k) { B2[k] += (float)CS * B1[k] + s2[k]; B1[k] += s1[k]; }
  }
}

__global__ void k_vp(const float* __restrict__ x, const float* __restrict__ act,
                     const float* __restrict__ ws, float* __restrict__ out) {
  int c = blockIdx.x * blockDim.x + threadIdx.x;
  if (c >= CHUNKS) return;
  float B1x=ws[OFF_B1+c*3], B1y=ws[OFF_B1+c*3+1], B1z=ws[OFF_B1+c*3+2];
  float B2x=ws[OFF_B2+c*3], B2y=ws[OFF_B2+c*3+1], B2z=ws[OFF_B2+c*3+2];
  float p0x=x[0], p0y=x[1], p0z=x[2], v0x=x[3], v0y=x[4], v0z=x[5];
  float s1x=0,s1y=0,s1z=0, s2x=0,s2y=0,s2z=0;
  for (int i = 0; i < CS; ++i) {
    int t = c*CS + i;
    float c0,c1,c2; get_col2(x, out, t, c0, c1, c2);
    float fz = act[t*4];
    s1x += fz*c0; s1y += fz*c1; s1z += fz*c2 - GACC;
    // v_{t+1} = v0 + DT * S1_t
    out[t*18+3] = v0x + DTF*(B1x + s1x);
    out[t*18+4] = v0y + DTF*(B1y + s1y);
    out[t*18+5] = v0z + DTF*(B1z + s1z);
    // pos_{t+1} = pos0 + (t+1)*DT*v0 + DT^2 * S2_{t-1}
    float S2x = (i==0) ? B2x : (B2x + (float)i*B1x + s2x);
    float S2y = (i==0) ? B2y : (B2y + (float)i*B1y + s2y);
    float S2z = (i==0) ? B2z : (B2z + (float)i*B1z + s2z);
    float tp1 = (float)(t + 1) * DTF;
    out[t*18+0] = p0x + tp1*v0x + DTF*DTF*S2x;
    out[t*18+1] = p0y + tp1*v0y + DTF*DTF*S2y;
    out[t*18+2] = p0z + tp1*v0z + DTF*DTF*S2z;
    s2x += s1x; s2y += s1y; s2z += s1z;
  }
}

extern "C" void kernel_launch(void* const* d_in, const int* in_sizes, int n_in,
                              void* d_out, int out_size, void* d_ws, size_t ws_size,
                              hipStream_t stream) {
  const float* x   = (const float*)d_in[0];     // 18 floats
  const float* act = (const float*)d_in[1];     // 20000*4 floats
  float* out = (float*)d_out;                   // 360000 states + 80000 actions
  float* ws  = (float*)d_ws;

  k_copy_actions<<<(STEPS + 255)/256, 256, 0, stream>>>(
      (const float4*)act, (float4*)(out + STEPS*18), STEPS);
  k_chunk_wz <<<5, 32, 0, stream>>>(act, ws);
  k_scan_wz  <<<1, 1, 0, stream>>>(x, ws);
  k_chunk_aff<<<5, 32, 0, stream>>>(act, ws);
  k_scan_aff <<<1, 1, 0, stream>>>(x, ws);
  k_omega_E  <<<5, 32, 0, stream>>>(act, ws, out);
  k_scan_R   <<<1, 1, 0, stream>>>(x, ws);
  k_wmma_apply<<<STEPS/4, 32, 0, stream>>>(ws, out);
  k_dv       <<<5, 32, 0, stream>>>(x, act, out, ws);
  k_scan_vp  <<<1, 1, 0, stream>>>(ws);
  k_vp       <<<5, 32, 0, stream>>>(x, act, ws, out);
}